// GCNConv1_79946521247964
// MI455X (gfx1250) — compile-verified
//
#include <hip/hip_runtime.h>

#define N_NODES 100000
#define N_EDGES 3200000
#define D 128

typedef __attribute__((ext_vector_type(2))) float v2f;
typedef __attribute__((ext_vector_type(8))) float v8f;

// ---------------------------------------------------------------------------
// GEMM: n2n[m][n] = sum_k X0[m][k] * W[n][k] + b[n]
// One wave32 computes one 16x16 output tile via V_WMMA_F32_16X16X4_F32,
// looping K in steps of 4 (32 WMMAs). Block = 256 threads = 8 waves = 8 N-tiles
// (covers the full D_OUT=128), grid.x = M/16 = 6250 tiles.
// A-layout (fp32 16x4): lanes 0-15 -> K = {k, k+1}, lanes 16-31 -> K = {k+2, k+3},
// row M = m0 + lane%16  => one contiguous float2 load per lane per step.
// B = W^T, column n of B is row n of W (row-major) => identical addressing.
// ---------------------------------------------------------------------------
__global__ void gcn_gemm_wmma(const float* __restrict__ X,
                              const float* __restrict__ W,
                              const float* __restrict__ bias,
                              float* __restrict__ out) {
    const int lane = threadIdx.x & 31;
    const int wave = threadIdx.x >> 5;   // 0..7 -> N tile
    const int half = lane >> 4;          // 0 or 1 (K sub-split)
    const int lid  = lane & 15;          // row/col within tile
    const int m0 = blockIdx.x * 16;
    const int n0 = wave * 16;

    const float* arow = X + (size_t)(m0 + lid) * D;   // A row (M = m0+lid)
    const float* brow = W + (size_t)(n0 + lid) * D;   // B col n0+lid == W row n0+lid

    v8f acc = {};
    #pragma unroll
    for (int k = 0; k < D; k += 4) {
        v2f a = *(const v2f*)(arow + k + 2 * half);
        v2f b = *(const v2f*)(brow + k + 2 * half);
        // 8 args: (neg_a, A, neg_b, B, c_mod, C, reuse_a, reuse_b)
        acc = __builtin_amdgcn_wmma_f32_16x16x4_f32(
            false, a, false, b, (short)0, acc, false, false);
    }

    const float bv = bias[n0 + lid];
    float* ocol = out + n0 + lid;
    #pragma unroll
    for (int v = 0; v < 8; ++v) {
        // C/D layout: VGPR v, lanes 0-15 -> M = m0+v, lanes 16-31 -> M = m0+v+8
        ocol[(size_t)(m0 + v + 8 * half) * D] = acc[v] + bv;
    }
}

// ---------------------------------------------------------------------------
// Zero the accumulator/output buffer (harness poisons d_out; atomics need 0).
// ---------------------------------------------------------------------------
__global__ void gcn_zero(float4* __restrict__ out, int n4) {
    int i = blockIdx.x * blockDim.x + threadIdx.x;
    if (i < n4) out[i] = make_float4(0.f, 0.f, 0.f, 0.f);
}

// ---------------------------------------------------------------------------
// Edge scatter: out[row,:] += n2n[col,:] * val. 128 threads per edge, one
// float per thread: the 512B row gather is fully coalesced and (51.2 MB n2n,
// 51.2 MB out) both live in the 192 MB L2, so gathers + f32 atomics stay
// on-die. global_atomic_add_f32 without return (STOREcnt path).
// ---------------------------------------------------------------------------
__global__ void gcn_scatter(const float* __restrict__ n2n,
                            const float* __restrict__ edge_val,
                            const int*   __restrict__ edge_row,
                            const int*   __restrict__ edge_col,
                            float* __restrict__ out) {
    long long t = (long long)blockIdx.x * blockDim.x + threadIdx.x;
    int e = (int)(t >> 7);     // 128 threads per edge
    int f = (int)(t & 127);
    if (e < N_EDGES) {
        const float v  = edge_val[e];
        const int col  = edge_col[e];
        const int row  = edge_row[e];
        const float msg = n2n[(size_t)col * D + f] * v;
        atomicAdd(out + (size_t)row * D + f, msg);
    }
}

// ---------------------------------------------------------------------------
// ReLU epilogue, vectorized float4 (can't fuse into atomic accumulation).
// ---------------------------------------------------------------------------
__global__ void gcn_relu(float4* __restrict__ out, int n4) {
    int i = blockIdx.x * blockDim.x + threadIdx.x;
    if (i < n4) {
        float4 x = out[i];
        x.x = fmaxf(x.x, 0.f);
        x.y = fmaxf(x.y, 0.f);
        x.z = fmaxf(x.z, 0.f);
        x.w = fmaxf(x.w, 0.f);
        out[i] = x;
    }
}

extern "C" void kernel_launch(void* const* d_in, const int* in_sizes, int n_in,
                              void* d_out, int out_size, void* d_ws, size_t ws_size,
                              hipStream_t stream) {
    const float* X0       = (const float*)d_in[0];   // [100000,128]
    const float* W        = (const float*)d_in[1];   // [128,128]
    const float* b        = (const float*)d_in[2];   // [128]
    const float* edge_val = (const float*)d_in[3];   // [3.2M]
    const int*   edge_row = (const int*)d_in[4];     // [3.2M]
    const int*   edge_col = (const int*)d_in[5];     // [3.2M]
    float* out = (float*)d_out;                      // [100000,128]
    float* n2n = (float*)d_ws;                       // 51.2 MB scratch

    // 1) Dense GEMM via fp32 WMMA: 6250 M-tiles x (8 waves = full N)
    gcn_gemm_wmma<<<N_NODES / 16, 256, 0, stream>>>(X0, W, b, n2n);

    // 2) Zero accumulator
    const int n4 = N_NODES * D / 4;  // 3.2M float4
    gcn_zero<<<(n4 + 255) / 256, 256, 0, stream>>>((float4*)out, n4);

    // 3) Sparse scatter with f32 atomics (dominant phase, L2-resident)
    const long long scatter_threads = (long long)N_EDGES * 128;
    const int scatter_blocks = (int)((scatter_threads + 255) / 256);
    gcn_scatter<<<scatter_blocks, 256, 0, stream>>>(n2n, edge_val, edge_row, edge_col, out);

    // 4) ReLU
    gcn_relu<<<(n4 + 255) / 256, 256, 0, stream>>>((float4*)out, n4);
}